// ResidualGINEGATBlock_29240137351457
// MI455X (gfx1250) — compile-verified
//
#include <hip/hip_runtime.h>
#include <hip/hip_bf16.h>

typedef float v2f __attribute__((ext_vector_type(2)));
typedef float v8f __attribute__((ext_vector_type(8)));

#define DIMK 128
#define WAVES_PER_BLOCK 8

// ---------------------------------------------------------------------------
// utility
// ---------------------------------------------------------------------------
__global__ void zero_kernel(float* __restrict__ p, size_t nfloats) {
    size_t i = (size_t)blockIdx.x * blockDim.x + threadIdx.x;
    size_t stride = (size_t)gridDim.x * blockDim.x;
    for (; i < nfloats; i += stride) p[i] = 0.0f;
}

__device__ __forceinline__ float lrelu02(float v) { return v > 0.0f ? v : 0.2f * v; }

__device__ __forceinline__ void atomicMaxF(float* addr, float v) {
    // ordered-int trick: valid for mixed signs
    if (v >= 0.0f) atomicMax((int*)addr, __float_as_int(v));
    else           atomicMin((unsigned int*)addr, __float_as_uint(v));
}

// ---------------------------------------------------------------------------
// GINE edge pass: agg[dst] += relu(x[src] + edge_attr[e]); one wave per edge
// ---------------------------------------------------------------------------
__global__ void gine_edge_kernel(const float* __restrict__ x,
                                 const float* __restrict__ ea,
                                 const int* __restrict__ src,
                                 const int* __restrict__ dst,
                                 float* __restrict__ agg, int E) {
    int e = blockIdx.x * WAVES_PER_BLOCK + (threadIdx.x >> 5);
    if (e >= E) return;
    int lane = threadIdx.x & 31;
    int s = src[e], d = dst[e];
    const float4 xv = *(const float4*)(x  + (size_t)s * DIMK + lane * 4);
    const float4 ev = *(const float4*)(ea + (size_t)e * DIMK + lane * 4);
    float m0 = fmaxf(xv.x + ev.x, 0.0f);
    float m1 = fmaxf(xv.y + ev.y, 0.0f);
    float m2 = fmaxf(xv.z + ev.z, 0.0f);
    float m3 = fmaxf(xv.w + ev.w, 0.0f);
    float* ap = agg + (size_t)d * DIMK + lane * 4;
    atomicAdd(ap + 0, m0);
    atomicAdd(ap + 1, m1);
    atomicAdd(ap + 2, m2);
    atomicAdd(ap + 3, m3);
}

// ---------------------------------------------------------------------------
// WMMA GEMM: out[M,128] = act(A[M,128] @ W[128,128] + bias)
//   FUSE: A = (1+eps)*x + agg computed on the fly (GINE input fusion)
// Block = 256 thr (8 waves); block tile = 128 rows x 128 cols; wave = 16 rows.
// W staged in 64KB LDS *pre-swizzled* so each B fragment's K-pair is a single
// aligned ds_load_b64. K=128 consumed as 32 steps of V_WMMA_F32_16X16X4_F32.
// K-loop pinned to unroll 1 (no spills); A stream software-pipelined with
// raw-load-now / fma-at-consume so the global latency hides behind the WMMAs.
// ---------------------------------------------------------------------------
template<bool FUSE, bool RELU, bool HASBIAS>
__global__ void gemm128_wmma(const float* __restrict__ Ain,
                             const float* __restrict__ x,
                             const float* __restrict__ agg,
                             const float* __restrict__ epsp,
                             const float* __restrict__ W,
                             const float* __restrict__ bias,
                             float* __restrict__ out,
                             int nrows) {
    // swizzled layout: lW[((k>>2)*2 + ((k>>1)&1))*256 + n*2 + (k&1)] = W[k][n]
    // -> fragment pair (W[k0][n], W[k0+1][n]) with k0 = kk*4 + hi*2 is the
    //    contiguous float2 at lW + (kk*2+hi)*256 + n*2.
    __shared__ float lW[DIMK * DIMK];   // 64 KB
    int tid = threadIdx.x;
#pragma unroll
    for (int i = 0; i < (DIMK * DIMK) / 256; ++i) {
        int idx = i * 256 + tid;
        int k = idx >> 7, n = idx & 127;
        int dstidx = (((k >> 2) << 1) + ((k >> 1) & 1)) * 256 + (n << 1) + (k & 1);
        lW[dstidx] = W[idx];
    }
    __syncthreads();

    int wave = tid >> 5;
    int lane = tid & 31;
    int hi   = lane >> 4;          // lane half (K split)
    int ml   = lane & 15;
    int row0 = blockIdx.x * 128 + wave * 16;

    float epsv = 0.0f;
    if (FUSE) epsv = 1.0f + epsp[0];

    v8f acc[8];
    const v8f vzero = {0.f, 0.f, 0.f, 0.f, 0.f, 0.f, 0.f, 0.f};
#pragma unroll
    for (int t = 0; t < 8; ++t) acc[t] = vzero;

    // clamp OOB rows instead of branching: their D rows are never stored
    int arow = row0 + ml;
    if (arow > nrows - 1) arow = nrows - 1;
    size_t abase = (size_t)arow * DIMK + hi * 2;

    const float* lWf = lW + hi * 256 + ml * 2;   // this lane's fragment base

    // software pipeline: raw loads issued one iteration ahead; the fused
    // combine happens at consume time (after next loads are issued) so the
    // compiler can use a partial loadcnt wait and overlap with the WMMAs.
    float2 xr_n = {0.f, 0.f}, ar_n;
    if (FUSE) {
        xr_n = *(const float2*)(x   + abase);
        ar_n = *(const float2*)(agg + abase);
    } else {
        ar_n = *(const float2*)(Ain + abase);
    }
#pragma unroll 1
    for (int kk = 0; kk < 32; ++kk) {
        float2 xr = xr_n, ar = ar_n;
        int kn = kk < 31 ? kk + 1 : 31;          // scalar clamp, branchless
        if (FUSE) {
            xr_n = *(const float2*)(x   + abase + kn * 4);
            ar_n = *(const float2*)(agg + abase + kn * 4);
        } else {
            ar_n = *(const float2*)(Ain + abase + kn * 4);
        }
        v2f a;
        if (FUSE) {
            a.x = epsv * xr.x + ar.x;
            a.y = epsv * xr.y + ar.y;
        } else {
            a.x = ar.x; a.y = ar.y;
        }
        const float* bp = lWf + kk * 512;        // (kk*2+hi)*256 + ml*2
#pragma unroll
        for (int t = 0; t < 8; ++t) {
            v2f b = *(const v2f*)(bp + t * 32);  // single aligned ds_load_b64
            acc[t] = __builtin_amdgcn_wmma_f32_16x16x4_f32(
                false, a, false, b, (short)0, acc[t], false, false);
        }
    }

    // epilogue: C/D layout — VGPR j holds row j (lanes 0-15) / row j+8 (16-31)
#pragma unroll
    for (int j = 0; j < 8; ++j) {
        int r = row0 + j + hi * 8;
        if (r < nrows) {
#pragma unroll
            for (int t = 0; t < 8; ++t) {
                int col = t * 16 + ml;
                float v = acc[t][j];
                if (HASBIAS) v += bias[col];
                if (RELU) v = fmaxf(v, 0.0f);
                out[(size_t)r * DIMK + col] = v;
            }
        }
    }
}

// ---------------------------------------------------------------------------
// LayerNorm(+opt bias, relu) + residual; one wave per row (4 f32/lane)
// ---------------------------------------------------------------------------
__global__ void ln_kernel(const float* __restrict__ in,
                          const float* __restrict__ bias,
                          const float* __restrict__ res,
                          const float* __restrict__ gam,
                          const float* __restrict__ bet,
                          float* __restrict__ out, int n) {
    int node = blockIdx.x * WAVES_PER_BLOCK + (threadIdx.x >> 5);
    if (node >= n) return;
    int lane = threadIdx.x & 31;
    size_t base = (size_t)node * DIMK + lane * 4;
    float4 v = *(const float4*)(in + base);
    if (bias) {
        const float4 bb = *(const float4*)(bias + lane * 4);
        v.x += bb.x; v.y += bb.y; v.z += bb.z; v.w += bb.w;
    }
    v.x = fmaxf(v.x, 0.0f); v.y = fmaxf(v.y, 0.0f);
    v.z = fmaxf(v.z, 0.0f); v.w = fmaxf(v.w, 0.0f);
    float s = v.x + v.y + v.z + v.w;
    float q = v.x * v.x + v.y * v.y + v.z * v.z + v.w * v.w;
#pragma unroll
    for (int off = 1; off < 32; off <<= 1) {
        s += __shfl_xor(s, off, 32);
        q += __shfl_xor(q, off, 32);
    }
    float mean = s * (1.0f / 128.0f);
    float var  = q * (1.0f / 128.0f) - mean * mean;
    float rstd = rsqrtf(var + 1e-5f);
    float4 g4 = *(const float4*)(gam + lane * 4);
    float4 b4 = *(const float4*)(bet + lane * 4);
    float4 r4 = *(const float4*)(res + base);
    float4 o;
    o.x = (v.x - mean) * rstd * g4.x + b4.x + r4.x;
    o.y = (v.y - mean) * rstd * g4.y + b4.y + r4.y;
    o.z = (v.z - mean) * rstd * g4.z + b4.z + r4.z;
    o.w = (v.w - mean) * rstd * g4.w + b4.w + r4.w;
    *(float4*)(out + base) = o;
}

// ---------------------------------------------------------------------------
// GAT attention coefficients + self-loop max init; one wave per node
// ---------------------------------------------------------------------------
__global__ void att_kernel(const float* __restrict__ xp,
                           const float* __restrict__ atts,
                           const float* __restrict__ attd,
                           float* __restrict__ a_src,
                           float* __restrict__ a_dst,
                           float* __restrict__ mbuf, int n) {
    int node = blockIdx.x * WAVES_PER_BLOCK + (threadIdx.x >> 5);
    if (node >= n) return;
    int lane = threadIdx.x & 31;
    float4 v  = *(const float4*)(xp + (size_t)node * DIMK + lane * 4);
    float4 as = *(const float4*)(atts + lane * 4);
    float4 ad = *(const float4*)(attd + lane * 4);
    float ps = v.x * as.x + v.y * as.y + v.z * as.z + v.w * as.w;
    float pd = v.x * ad.x + v.y * ad.y + v.z * ad.z + v.w * ad.w;
    // reduce within groups of 8 lanes = one head (32 dims)
#pragma unroll
    for (int off = 1; off < 8; off <<= 1) {
        ps += __shfl_xor(ps, off, 32);
        pd += __shfl_xor(pd, off, 32);
    }
    if ((lane & 7) == 0) {
        int h = lane >> 3;
        a_src[(size_t)node * 4 + h] = ps;
        a_dst[(size_t)node * 4 + h] = pd;
        mbuf [(size_t)node * 4 + h] = lrelu02(ps + pd);   // self-loop logit
    }
}

// edge atomic-max over logits; one thread per (edge, head)
__global__ void edge_max_kernel(const float* __restrict__ a_src,
                                const float* __restrict__ a_dst,
                                const int* __restrict__ src,
                                const int* __restrict__ dst,
                                float* __restrict__ mbuf, int EH) {
    int t = blockIdx.x * blockDim.x + threadIdx.x;
    if (t >= EH) return;
    int e = t >> 2, h = t & 3;
    float l = lrelu02(a_src[(size_t)src[e] * 4 + h] + a_dst[(size_t)dst[e] * 4 + h]);
    atomicMaxF(&mbuf[(size_t)dst[e] * 4 + h], l);
}

// s[i,h] = exp(self_logit - m); one thread per (node, head)
__global__ void sinit_kernel(const float* __restrict__ a_src,
                             const float* __restrict__ a_dst,
                             const float* __restrict__ mbuf,
                             float* __restrict__ sbuf, int NH) {
    int t = blockIdx.x * blockDim.x + threadIdx.x;
    if (t >= NH) return;
    float l = lrelu02(a_src[t] + a_dst[t]);
    sbuf[t] = __expf(l - mbuf[t]);
}

// edge exp-sum; one thread per (edge, head)
__global__ void edge_expsum_kernel(const float* __restrict__ a_src,
                                   const float* __restrict__ a_dst,
                                   const int* __restrict__ src,
                                   const int* __restrict__ dst,
                                   const float* __restrict__ mbuf,
                                   float* __restrict__ sbuf, int EH) {
    int t = blockIdx.x * blockDim.x + threadIdx.x;
    if (t >= EH) return;
    int e = t >> 2, h = t & 3;
    size_t di = (size_t)dst[e] * 4 + h;
    float l = lrelu02(a_src[(size_t)src[e] * 4 + h] + a_dst[di]);
    atomicAdd(&sbuf[di], __expf(l - mbuf[di]));
}

// out[i] = alpha_self * xp[i]  (initializes accumulator); one wave per node
__global__ void selfagg_kernel(const float* __restrict__ xp,
                               const float* __restrict__ a_src,
                               const float* __restrict__ a_dst,
                               const float* __restrict__ mbuf,
                               const float* __restrict__ sbuf,
                               float* __restrict__ outacc, int n) {
    int node = blockIdx.x * WAVES_PER_BLOCK + (threadIdx.x >> 5);
    if (node >= n) return;
    int lane = threadIdx.x & 31;
    int h = lane >> 3;                 // 32 dims/head, 4 dims/lane
    size_t ih = (size_t)node * 4 + h;
    float l = lrelu02(a_src[ih] + a_dst[ih]);
    float alpha = __expf(l - mbuf[ih]) / (sbuf[ih] + 1e-16f);
    size_t base = (size_t)node * DIMK + lane * 4;
    float4 v = *(const float4*)(xp + base);
    float4 o;
    o.x = alpha * v.x; o.y = alpha * v.y; o.z = alpha * v.z; o.w = alpha * v.w;
    *(float4*)(outacc + base) = o;
}

// out[dst] += alpha_e * xp[src]; one wave per edge
__global__ void edge_agg_kernel(const float* __restrict__ xp,
                                const float* __restrict__ a_src,
                                const float* __restrict__ a_dst,
                                const int* __restrict__ src,
                                const int* __restrict__ dst,
                                const float* __restrict__ mbuf,
                                const float* __restrict__ sbuf,
                                float* __restrict__ outacc, int E) {
    int e = blockIdx.x * WAVES_PER_BLOCK + (threadIdx.x >> 5);
    if (e >= E) return;
    int lane = threadIdx.x & 31;
    int sI = src[e], dI = dst[e];
    int h = lane >> 3;
    size_t dh = (size_t)dI * 4 + h;
    float l = lrelu02(a_src[(size_t)sI * 4 + h] + a_dst[dh]);
    float alpha = __expf(l - mbuf[dh]) / (sbuf[dh] + 1e-16f);
    float4 v = *(const float4*)(xp + (size_t)sI * DIMK + lane * 4);
    float* op = outacc + (size_t)dI * DIMK + lane * 4;
    atomicAdd(op + 0, alpha * v.x);
    atomicAdd(op + 1, alpha * v.y);
    atomicAdd(op + 2, alpha * v.z);
    atomicAdd(op + 3, alpha * v.w);
}

// ---------------------------------------------------------------------------
// launcher
// ---------------------------------------------------------------------------
extern "C" void kernel_launch(void* const* d_in, const int* in_sizes, int n_in,
                              void* d_out, int out_size, void* d_ws, size_t ws_size,
                              hipStream_t stream) {
    const float* x    = (const float*)d_in[0];
    const int*   ei   = (const int*)  d_in[1];
    const float* ea   = (const float*)d_in[2];
    const float* epsp = (const float*)d_in[3];
    const float* w1   = (const float*)d_in[4];
    const float* b1   = (const float*)d_in[5];
    const float* w2   = (const float*)d_in[6];
    const float* b2   = (const float*)d_in[7];
    const float* g1   = (const float*)d_in[8];
    const float* be1  = (const float*)d_in[9];
    const float* gw   = (const float*)d_in[10];
    const float* atts = (const float*)d_in[11];
    const float* attd = (const float*)d_in[12];
    const float* gb   = (const float*)d_in[13];
    const float* g2   = (const float*)d_in[14];
    const float* be2  = (const float*)d_in[15];

    const int N = in_sizes[0] / DIMK;
    const int E = in_sizes[1] / 2;
    const int* src = ei;
    const int* dst = ei + E;

    float* ws = (float*)d_ws;
    const size_t NL = (size_t)N * DIMK;
    // buffer plan (~160MB): buf0 = agg -> xp (agg dead after GEMM1);
    //                       buf1 = mlp-hidden -> GAT accumulator;
    //                       buf2 = h (persists through GAT)
    float* buf0 = ws;              // agg, later xp
    float* buf1 = ws + NL;         // t1, later GAT out accumulator
    float* hbuf = ws + 2 * NL;     // h
    float* asr  = ws + 3 * NL;
    float* adt  = asr + (size_t)N * 4;
    float* mbuf = adt + (size_t)N * 4;
    float* sbuf = mbuf + (size_t)N * 4;
    float* outp = (float*)d_out;

    const int nodeWaveBlocks = (N + WAVES_PER_BLOCK - 1) / WAVES_PER_BLOCK;
    const int edgeWaveBlocks = (E + WAVES_PER_BLOCK - 1) / WAVES_PER_BLOCK;
    const int ehBlocks = (4 * E + 255) / 256;
    const int nhBlocks = (4 * N + 255) / 256;
    const int gemmBlocks = (N + 127) / 128;

    // 1) zero agg
    zero_kernel<<<2048, 256, 0, stream>>>(buf0, NL);
    // 2) GINE edge scatter
    gine_edge_kernel<<<edgeWaveBlocks, 256, 0, stream>>>(x, ea, src, dst, buf0, E);
    // 3) t1 = relu(((1+eps)x + agg) @ W1 + b1)
    gemm128_wmma<true, true, true><<<gemmBlocks, 256, 0, stream>>>(
        nullptr, x, buf0, epsp, w1, b1, buf1, N);
    // 4) h2 = t1 @ W2 + b2
    gemm128_wmma<false, false, true><<<gemmBlocks, 256, 0, stream>>>(
        buf1, nullptr, nullptr, nullptr, w2, b2, hbuf, N);
    // 5) h = LN(relu(h2)) + x   (in place)
    ln_kernel<<<nodeWaveBlocks, 256, 0, stream>>>(hbuf, nullptr, x, g1, be1, hbuf, N);
    // 6) xp = h @ gat_w
    gemm128_wmma<false, false, false><<<gemmBlocks, 256, 0, stream>>>(
        hbuf, nullptr, nullptr, nullptr, gw, nullptr, buf0, N);
    // 7) attention coefficients + self-loop max init
    att_kernel<<<nodeWaveBlocks, 256, 0, stream>>>(buf0, atts, attd, asr, adt, mbuf, N);
    // 8) edge atomic max
    edge_max_kernel<<<ehBlocks, 256, 0, stream>>>(asr, adt, src, dst, mbuf, 4 * E);
    // 9) s init (self-loop term)
    sinit_kernel<<<nhBlocks, 256, 0, stream>>>(asr, adt, mbuf, sbuf, 4 * N);
    // 10) edge exp-sum
    edge_expsum_kernel<<<ehBlocks, 256, 0, stream>>>(asr, adt, src, dst, mbuf, sbuf, 4 * E);
    // 11) out = alpha_self * xp (initializes accumulator)
    selfagg_kernel<<<nodeWaveBlocks, 256, 0, stream>>>(buf0, asr, adt, mbuf, sbuf, buf1, N);
    // 12) out[dst] += alpha_e * xp[src]
    edge_agg_kernel<<<edgeWaveBlocks, 256, 0, stream>>>(buf0, asr, adt, src, dst, mbuf, sbuf, buf1, E);
    // 13) g = LN(relu(out + gat_b)) + h  -> d_out
    ln_kernel<<<nodeWaveBlocks, 256, 0, stream>>>(buf1, gb, hbuf, g2, be2, outp, N);
}